// RetinaNetDetector_2482491097100
// MI455X (gfx1250) — compile-verified
//
#include <hip/hip_runtime.h>
#include <hip/hip_bf16.h>

// ---------------- problem constants (from reference) ----------------
#define NB    4
#define NC    8
#define NA    9
#define NLVL  5
#define TOPK  1000
#define NPROP (NLVL * TOPK)   // 5000 candidates per image
#define DETS  300
#define IMGSZ 512.0f
#define SCORE_TH 0.05f
#define IOU_TH   0.5f
#define BBOX_CLIP 4.135166556742356f   // log(1000/16)

// ---------------- tuning ----------------
#define CHUNK 4096      // floats per TDM chunk (16KB) -> double buffered
#define CAND  2048      // gather buffer (>= TOPK + tie slack)
#define HB    2048      // histogram bins
#define NPAD  8192      // padded sort size for 5000 candidates

#if defined(__has_builtin)
#if __has_builtin(__builtin_amdgcn_tensor_load_to_lds) && __has_builtin(__builtin_amdgcn_s_wait_tensorcnt)
#define USE_TDM 1
#endif
#endif
#ifndef USE_TDM
#define USE_TDM 0
#endif

typedef unsigned int u32x4 __attribute__((ext_vector_type(4)));
typedef int          i32x8 __attribute__((ext_vector_type(8)));
typedef int          i32x4 __attribute__((ext_vector_type(4)));

// monotone float->uint key (order-preserving for all finite floats)
__device__ __forceinline__ unsigned fkey(float f) {
  unsigned u = __float_as_uint(f);
  return (u & 0x80000000u) ? ~u : (u | 0x80000000u);
}
__device__ __forceinline__ float fdekey(unsigned k) {
  unsigned u = (k & 0x80000000u) ? (k & 0x7FFFFFFFu) : ~k;
  return __uint_as_float(u);
}

#if USE_TDM
// Build a 2D TDM descriptor for a contiguous chunk of `valid` f32 elements:
// tile = CHUNK x 1, tensor_dim0 = valid, so the OOB tail returns zeros.
// Layout per CDNA5 ISA ch.8 (group0 128b, group1 256b; groups 2/3 zero for 2D).
// 6-arg builtin form (clang-23 / therock-10.0 headers).
__device__ __forceinline__ void tdm_load_chunk(float* ldsdst, const float* gsrc,
                                               unsigned valid) {
  unsigned long long ga = (unsigned long long)(const void*)gsrc;
  unsigned lds = (unsigned)(unsigned long long)(void*)ldsdst; // LDS byte offset = addr[31:0]
  u32x4 g0;
  g0[0] = 1u;                                            // count=1, user mode
  g0[1] = lds;                                           // lds_addr
  g0[2] = (unsigned)(ga & 0xFFFFFFFFull);                // global_addr[31:0]
  g0[3] = (unsigned)((ga >> 32) & 0x1FFFFFFull) | (2u << 30); // addr[56:32] | type=2
  unsigned td0 = valid;                                  // tensor_dim0 (elements)
  i32x8 g1;
  g1[0] = (int)(2u << 16);                               // wg_mask=0 | data_size=4B
  g1[1] = (int)((td0 & 0xFFFFu) << 16);                  // abar_addr=0 | td0[15:0]
  g1[2] = (int)((td0 >> 16) | (1u << 16));               // td0[31:16] | tdim1[15:0]=1
  g1[3] = (int)(((unsigned)CHUNK) << 16);                // tdim1[31:16]=0 | tile_dim0
  g1[4] = 1;                                             // tile_dim1=1 | tile_dim2=0
  g1[5] = (int)td0;                                      // stride0[31:0]
  g1[6] = 0;                                             // stride0[47:32] | stride1[15:0]
  g1[7] = 0;                                             // stride1[47:16]
  i32x4 z4 = {0, 0, 0, 0};
  i32x8 z8 = {0, 0, 0, 0, 0, 0, 0, 0};
  __builtin_amdgcn_tensor_load_to_lds(g0, g1, z4, z4, z8, 0);
}
#endif

// Stream n contiguous floats through double-buffered LDS chunks; call fn(idx, val).
template <typename F>
__device__ void chunked_pass(const float* __restrict__ gbase, int n,
                             float* buf0, float* buf1, F fn) {
  float* bufs[2] = {buf0, buf1};
#if USE_TDM
  if (threadIdx.x == 0) tdm_load_chunk(buf0, gbase, (unsigned)n);
#endif
  int it = 0;
  for (int c0 = 0; c0 < n; c0 += CHUNK, ++it) {
    float* cur = bufs[it & 1];
    __syncthreads();  // everyone done with the buffer we are about to refill
#if USE_TDM
    if (threadIdx.x == 0) {
      if (c0 + CHUNK < n) {
        tdm_load_chunk(bufs[(it + 1) & 1], gbase + c0 + CHUNK,
                       (unsigned)(n - c0 - CHUNK));
        __builtin_amdgcn_s_wait_tensorcnt(1);  // chunk `it` done (in-order)
      } else {
        __builtin_amdgcn_s_wait_tensorcnt(0);  // drain on last chunk
      }
    }
#else
    for (int i = threadIdx.x; i < CHUNK; i += blockDim.x) {
      int g = c0 + i;
      cur[i] = (g < n) ? gbase[g] : 0.0f;
    }
#endif
    __syncthreads();  // chunk `it` resident in LDS
    int lim = (n - c0 < CHUNK) ? (n - c0) : CHUNK;
    for (int i = threadIdx.x; i < lim; i += blockDim.x) fn(c0 + i, cur[i]);
  }
  __syncthreads();
}

// ---------------- Kernel 1: per-level sigmoid + top-1000 + box decode ----------------
__global__ __launch_bounds__(256) void retina_topk_kernel(
    const float* __restrict__ cls, const float* __restrict__ reg,
    const float* __restrict__ anchors, int f, int lvl,
    float* __restrict__ cboxes, float* __restrict__ cscores,
    int* __restrict__ clabels) {
  __shared__ float bufA[CHUNK], bufB[CHUNK];
  __shared__ unsigned hist[HB];
  __shared__ unsigned ckey[CAND], cidx[CAND];
  __shared__ unsigned s_cnt, s_bcut, s_above, s_base;

  const int b = blockIdx.x;
  const int tid = threadIdx.x;
  const int ff = f * f;
  const int n = NC * NA * ff;                 // total scores this (image, level)
  const float* gcls = cls + (size_t)b * n;

  // ---- Phase A: coarse histogram over key[31:21] ----
  for (int i = tid; i < HB; i += 256) hist[i] = 0;
  __syncthreads();
  chunked_pass(gcls, n, bufA, bufB,
               [&](int, float v) { atomicAdd(&hist[fkey(v) >> 21], 1u); });
  if (tid == 0) {
    unsigned cum = 0, above = 0; int bc = 0;
    for (int bi = HB - 1; bi >= 0; --bi) {
      if (cum + hist[bi] >= TOPK) { bc = bi; above = cum; break; }
      cum += hist[bi];
    }
    s_bcut = (unsigned)bc; s_above = above;
  }
  __syncthreads();

  // ---- Phase B: refine within cutoff bin over key[20:10] ----
  unsigned bcut = s_bcut;
  for (int i = tid; i < HB; i += 256) hist[i] = 0;
  __syncthreads();
  chunked_pass(gcls, n, bufA, bufB, [&](int, float v) {
    unsigned k = fkey(v);
    if ((k >> 21) == bcut) atomicAdd(&hist[(k >> 10) & 0x7FFu], 1u);
  });
  if (tid == 0) {
    unsigned need = TOPK - s_above;
    unsigned cum = 0; int bc2 = 0;
    for (int bi = HB - 1; bi >= 0; --bi) {
      cum += hist[bi];
      if (cum >= need) { bc2 = bi; break; }
    }
    s_base = (bcut << 21) | ((unsigned)bc2 << 10);
    s_cnt = 0;
  }
  __syncthreads();

  // ---- Phase C: gather all elements with key >= cutoff ----
  unsigned base = s_base;
  for (int i = tid; i < CAND; i += 256) { ckey[i] = 0; cidx[i] = 0; }
  __syncthreads();
  chunked_pass(gcls, n, bufA, bufB, [&](int r, float v) {
    unsigned k = fkey(v);
    if (k >= base) {
      unsigned p = atomicAdd(&s_cnt, 1u);
      if (p < CAND) { ckey[p] = k; cidx[p] = (unsigned)r; }
    }
  });
  __syncthreads();

  // ---- Phase D: bitonic sort CAND entries, descending by key ----
  for (unsigned kk = 2; kk <= CAND; kk <<= 1) {
    for (unsigned j = kk >> 1; j > 0; j >>= 1) {
      __syncthreads();
      for (unsigned i = tid; i < CAND; i += 256) {
        unsigned ixj = i ^ j;
        if (ixj > i) {
          unsigned ka = ckey[i], kb = ckey[ixj];
          bool up = ((i & kk) == 0);
          if (up ? (ka < kb) : (ka > kb)) {
            ckey[i] = kb; ckey[ixj] = ka;
            unsigned t = cidx[i]; cidx[i] = cidx[ixj]; cidx[ixj] = t;
          }
        }
      }
    }
  }
  __syncthreads();

  // ---- Phase E: decode top-1000, write candidates ----
  for (int k = tid; k < TOPK; k += 256) {
    int o = b * NPROP + lvl * TOPK + k;
    unsigned key = ckey[k];
    float bx0 = 0.f, by0 = 0.f, bx1 = 0.f, by1 = 0.f, sc = -1.0f;
    int lab = 0;
    if (key != 0u) {
      int r = (int)cidx[k];                 // raw index into [72][f][f]
      int x = r % f, y = (r / f) % f, chn = r / ff;
      int a = chn / NC; lab = chn % NC;
      float logit = fdekey(key);
      sc = 1.0f / (1.0f + __expf(-logit));  // sigmoid
      size_t rbase = (size_t)b * (4 * NA * ff) + (size_t)(a * 4) * ff +
                     (size_t)y * f + x;
      float dx = reg[rbase];
      float dy = reg[rbase + (size_t)ff];
      float dw = fminf(reg[rbase + 2 * (size_t)ff], BBOX_CLIP);
      float dh = fminf(reg[rbase + 3 * (size_t)ff], BBOX_CLIP);
      int ai = (y * f + x) * NA + a;
      float a0 = anchors[ai * 4 + 0], a1 = anchors[ai * 4 + 1];
      float a2 = anchors[ai * 4 + 2], a3 = anchors[ai * 4 + 3];
      float aw = a2 - a0, ah = a3 - a1;
      float acx = a0 + 0.5f * aw, acy = a1 + 0.5f * ah;
      float pcx = dx * aw + acx, pcy = dy * ah + acy;
      float pw = __expf(dw) * aw, ph = __expf(dh) * ah;
      bx0 = fminf(fmaxf(pcx - 0.5f * pw, 0.0f), IMGSZ);
      by0 = fminf(fmaxf(pcy - 0.5f * ph, 0.0f), IMGSZ);
      bx1 = fminf(fmaxf(pcx + 0.5f * pw, 0.0f), IMGSZ);
      by1 = fminf(fmaxf(pcy + 0.5f * ph, 0.0f), IMGSZ);
      if (!(sc > SCORE_TH)) sc = -1.0f;     // threshold -> invalid
    }
    cboxes[(size_t)o * 4 + 0] = bx0;
    cboxes[(size_t)o * 4 + 1] = by0;
    cboxes[(size_t)o * 4 + 2] = bx1;
    cboxes[(size_t)o * 4 + 3] = by1;
    cscores[o] = sc;
    clabels[o] = lab;
  }
}

// ---------------- Kernel 2: per-image sort + class-aware greedy NMS + top-300 ----------------
__global__ __launch_bounds__(1024) void retina_nms_kernel(
    const float* __restrict__ cboxes, const float* __restrict__ cscores,
    const int* __restrict__ clabels, float* __restrict__ oboxes,
    float* __restrict__ oscores, int* __restrict__ olabels) {
  __shared__ unsigned       skey[NPAD];
  __shared__ unsigned short sidx[NPAD];
  __shared__ unsigned char  alive[NPROP];

  const int b = blockIdx.x;
  const int tid = threadIdx.x;

  // default outputs (invalid): boxes 0, score 0, label -1
  for (int t = tid; t < DETS; t += 1024) {
    int o = b * DETS + t;
    oboxes[(size_t)o * 4 + 0] = 0.0f; oboxes[(size_t)o * 4 + 1] = 0.0f;
    oboxes[(size_t)o * 4 + 2] = 0.0f; oboxes[(size_t)o * 4 + 3] = 0.0f;
    oscores[o] = 0.0f; olabels[o] = -1;
  }

  for (int j = tid; j < NPAD; j += 1024) {
    if (j < NPROP) {
      skey[j] = fkey(cscores[b * NPROP + j]);
      sidx[j] = (unsigned short)j;
      __builtin_prefetch(&cboxes[((size_t)b * NPROP + j) * 4], 0, 3);
    } else { skey[j] = 0; sidx[j] = 0; }
  }

  // bitonic argsort, descending by score key
  for (unsigned kk = 2; kk <= NPAD; kk <<= 1) {
    for (unsigned j = kk >> 1; j > 0; j >>= 1) {
      __syncthreads();
      for (unsigned i = tid; i < NPAD; i += 1024) {
        unsigned ixj = i ^ j;
        if (ixj > i) {
          unsigned ka = skey[i], kb = skey[ixj];
          bool up = ((i & kk) == 0);
          if (up ? (ka < kb) : (ka > kb)) {
            skey[i] = kb; skey[ixj] = ka;
            unsigned short t2 = sidx[i]; sidx[i] = sidx[ixj]; sidx[ixj] = t2;
          }
        }
      }
    }
  }
  __syncthreads();

  // each thread caches its (up to 5) class-offset boxes in registers
  float jb[5][4]; int jvalid[5];
  for (int k = 0; k < 5; ++k) {
    int j = tid + k * 1024;
    jvalid[k] = (j < NPROP);
    if (j < NPROP) {
      alive[j] = (skey[j] > 0x80000000u) ? 1 : 0;  // score > 0
      int idx = sidx[j];
      const float* bp = &cboxes[((size_t)b * NPROP + idx) * 4];
      float off = (float)clabels[b * NPROP + idx] * (IMGSZ + 1.0f);
      jb[k][0] = bp[0] + off; jb[k][1] = bp[1] + off;
      jb[k][2] = bp[2] + off; jb[k][3] = bp[3] + off;
    } else {
      jb[k][0] = jb[k][1] = jb[k][2] = jb[k][3] = 0.0f;
    }
  }
  __syncthreads();

  // sequential greedy suppression (forward only, matches reference)
  for (int i = 0; i < NPROP; ++i) {
    __syncthreads();
    if (!alive[i]) continue;                       // uniform branch
    int idx = sidx[i];
    const float* bp = &cboxes[((size_t)b * NPROP + idx) * 4];
    float off = (float)clabels[b * NPROP + idx] * (IMGSZ + 1.0f);
    float ix0 = bp[0] + off, iy0 = bp[1] + off;
    float ix1 = bp[2] + off, iy1 = bp[3] + off;
    float iarea = (ix1 - ix0) * (iy1 - iy0);
    for (int k = 0; k < 5; ++k) {
      int j = tid + k * 1024;
      if (jvalid[k] && j > i && alive[j]) {
        float lx = fmaxf(ix0, jb[k][0]), ly = fmaxf(iy0, jb[k][1]);
        float rx = fminf(ix1, jb[k][2]), ry = fminf(iy1, jb[k][3]);
        float w = fmaxf(rx - lx, 0.0f), h = fmaxf(ry - ly, 0.0f);
        float inter = w * h;
        float ja = (jb[k][2] - jb[k][0]) * (jb[k][3] - jb[k][1]);
        float uni = fmaxf(iarea + ja - inter, 1e-7f);
        if (inter > IOU_TH * uni) alive[j] = 0;
      }
    }
  }
  __syncthreads();

  // compact first 300 survivors in score order
  if (tid == 0) {
    int cnt = 0;
    for (int j = 0; j < NPROP && cnt < DETS; ++j) {
      if (alive[j]) {
        int idx = sidx[j];
        int o = b * DETS + cnt;
        const float* bp = &cboxes[((size_t)b * NPROP + idx) * 4];
        oboxes[(size_t)o * 4 + 0] = bp[0];
        oboxes[(size_t)o * 4 + 1] = bp[1];
        oboxes[(size_t)o * 4 + 2] = bp[2];
        oboxes[(size_t)o * 4 + 3] = bp[3];
        oscores[o] = cscores[b * NPROP + idx];
        olabels[o] = clabels[b * NPROP + idx];
        ++cnt;
      }
    }
  }
}

// ---------------- host launcher ----------------
extern "C" void kernel_launch(void* const* d_in, const int* in_sizes, int n_in,
                              void* d_out, int out_size, void* d_ws, size_t ws_size,
                              hipStream_t stream) {
  (void)in_sizes; (void)n_in; (void)out_size; (void)ws_size;
  static const int FE[NLVL] = {64, 32, 16, 8, 4};

  // workspace: cand_boxes [B*5000*4] | cand_scores [B*5000] | cand_labels [B*5000]
  float* wsf = (float*)d_ws;
  float* cboxes  = wsf;                 // 80000 floats
  float* cscores = wsf + 80000;         // 20000 floats
  int*   clabels = (int*)(wsf + 100000);// 20000 ints

  for (int l = 0; l < NLVL; ++l) {
    const float* cls = (const float*)d_in[3 * l + 0];
    const float* reg = (const float*)d_in[3 * l + 1];
    const float* anc = (const float*)d_in[3 * l + 2];
    retina_topk_kernel<<<NB, 256, 0, stream>>>(cls, reg, anc, FE[l], l,
                                               cboxes, cscores, clabels);
  }

  float* ob = (float*)d_out;            // boxes  [B,300,4] = 4800 floats
  float* os = ob + NB * DETS * 4;       // scores [B,300]   = 1200 floats
  int*   ol = (int*)(ob + NB * DETS * 4 + NB * DETS); // labels [B,300]
  retina_nms_kernel<<<NB, 1024, 0, stream>>>(cboxes, cscores, clabels, ob, os, ol);
}